// AEKNNModel_58540404244550
// MI455X (gfx1250) — compile-verified
//
#include <hip/hip_runtime.h>
#include <hip/hip_bf16.h>
#include <math.h>

#define N_QUERY 1024
#define N_REF   100000
#define D       128
#define KTOP    8
#define NSPLIT  50
#define CHUNK   2000            // N_REF / NSPLIT (exact)
#define QTILE   16
#define NQTILES (N_QUERY / QTILE)   // 64
#define BLOCK   128
#define NWAVES  (BLOCK / 32)
#define ASTRIDE 130             // padded row stride (floats), 8B-aligned, bank-conflict free
#define INF_F   3.0e38f

typedef float v2f __attribute__((ext_vector_type(2)));
typedef float v8f __attribute__((ext_vector_type(8)));

// Streaming top-8 (smallest) keeper: unsorted 8 slots + cached max.
// Fast path = single compare; insert path replaces the current-max slot.
__device__ __forceinline__ void topk_insert(float (&s)[8], float& m, float d) {
    if (d < m) {
        bool done = false;
#pragma unroll
        for (int i = 0; i < 8; ++i) {
            if (!done && s[i] == m) { s[i] = d; done = true; }
        }
        float nm = s[0];
#pragma unroll
        for (int i = 1; i < 8; ++i) nm = fmaxf(nm, s[i]);
        m = nm;
    }
}

__global__ __launch_bounds__(BLOCK, 1) void knn_partial_kernel(
    const float* __restrict__ x,
    const float* __restrict__ base,
    float* __restrict__ part)           // [N_QUERY][NSPLIT][KTOP]
{
    __shared__ float As[QTILE * ASTRIDE];     // query tile, padded
    __shared__ float xns[QTILE];              // query norms
    __shared__ float cand[QTILE * 512];       // per-block candidate dump
    __shared__ float cand2[QTILE * 64];       // stage-1 reduced candidates

    const int split = blockIdx.x;             // ref chunk
    const int qt    = blockIdx.y;             // query tile
    const int tid   = threadIdx.x;
    const int wave  = tid >> 5;
    const int lane  = tid & 31;
    const int hi    = lane >> 4;              // lane-half: selects K parity / M-half
    const int l15   = lane & 15;

    const int q0 = qt * QTILE;
    const int r0 = split * CHUNK;

    // ---- stage query tile into LDS (coalesced) ----
    for (int i = tid; i < QTILE * D; i += BLOCK) {
        int r = i >> 7, c = i & 127;
        As[r * ASTRIDE + c] = x[(q0 + r) * D + c];
    }
    __syncthreads();

    // ---- query norms ----
    if (tid < QTILE) {
        float s = 0.f;
        for (int c = 0; c < D; ++c) { float v = As[tid * ASTRIDE + c]; s += v * v; }
        xns[tid] = s;
    }
    __syncthreads();

    // each lane needs xnorm for queries q = hi*8 + v, v = 0..7
    float xn[8];
#pragma unroll
    for (int v = 0; v < 8; ++v) xn[v] = xns[hi * 8 + v];

    // ---- hoist A fragments for all 32 K-slices into registers ----
    // A 16x4 fp32 layout per slice k: lanes 0-15 hold K=4k,4k+1 ; lanes 16-31 K=4k+2,4k+3
    const float* ap = &As[l15 * ASTRIDE + 2 * hi];
    v2f a[32];
#pragma unroll
    for (int k = 0; k < 32; ++k) a[k] = *(const v2f*)(ap + 4 * k);

    // per-lane running top-8 for 8 query streams
    float s8[8][8];
    float mx[8];
#pragma unroll
    for (int v = 0; v < 8; ++v) {
        mx[v] = INF_F;
#pragma unroll
        for (int i = 0; i < 8; ++i) s8[v][i] = INF_F;
    }

    // ---- main loop: 16x16 output tiles via V_WMMA_F32_16X16X4_F32 ----
    const int ntiles = CHUNK / 16;            // 125
    for (int t = wave; t < ntiles; t += NWAVES) {
        const int rr = r0 + t * 16 + l15;            // this lane's ref row (B: N=l15)
        const float* bp = base + (size_t)rr * D + 2 * hi;

        // prefetch next tile's rows for this lane (covers the 2 cachelines we use)
        if (t + NWAVES < ntiles) {
            const float* bpn = bp + (size_t)(16 * NWAVES) * D;
            __builtin_prefetch(bpn, 0, 3);
            __builtin_prefetch(bpn + 64, 0, 3);
        }

        // load the whole B tile (this lane's 32 dim-pairs) -> one deep load clause
        v2f b[32];
#pragma unroll
        for (int k = 0; k < 32; ++k) b[k] = *(const v2f*)(bp + 4 * k);

        // ref-norm partials: 4 independent accumulator chains
        float bn0 = 0.f, bn1 = 0.f, bn2 = 0.f, bn3 = 0.f;
#pragma unroll
        for (int k = 0; k < 32; k += 4) {
            bn0 += b[k + 0].x * b[k + 0].x + b[k + 0].y * b[k + 0].y;
            bn1 += b[k + 1].x * b[k + 1].x + b[k + 1].y * b[k + 1].y;
            bn2 += b[k + 2].x * b[k + 2].x + b[k + 2].y * b[k + 2].y;
            bn3 += b[k + 3].x * b[k + 3].x + b[k + 3].y * b[k + 3].y;
        }
        float bn = (bn0 + bn1) + (bn2 + bn3);

        // WMMA chain: two accumulators so back-to-back WMMAs don't serialize on C
        v8f c0 = {}, c1 = {};
#pragma unroll
        for (int k = 0; k < 32; k += 2) {
            c0 = __builtin_amdgcn_wmma_f32_16x16x4_f32(
                     false, a[k], false, b[k], (short)0, c0, false, false);
            c1 = __builtin_amdgcn_wmma_f32_16x16x4_f32(
                     false, a[k + 1], false, b[k + 1], (short)0, c1, false, false);
        }

        // lane l and l^16 hold complementary K-parity partial norms of the same ref
        bn += __shfl_xor(bn, 16, 32);

        // epilogue: distance + top-8 update. C layout: lane half gives M-half,
        // c[v] -> query q = hi*8 + v, ref = rr.
#pragma unroll
        for (int v = 0; v < 8; ++v) {
            float dot = c0[v] + c1[v];
            float d2  = xn[v] + bn - 2.0f * dot;
            float d   = sqrtf(fmaxf(d2, 0.f));
            topk_insert(s8[v], mx[v], d);
        }
    }

    // ---- dump per-lane lists: cand[q][wave*128 + l15*8 + i] ----
#pragma unroll
    for (int v = 0; v < 8; ++v) {
        int q = hi * 8 + v;
        float* dst = &cand[q * 512 + wave * 128 + l15 * 8];
#pragma unroll
        for (int i = 0; i < 8; ++i) dst[i] = s8[v][i];
    }
    __syncthreads();

    // ---- stage-1 reduce: 512 -> 64 per query (8 threads per query) ----
    {
        int q = tid >> 3, pp = tid & 7;
        const float* src = &cand[q * 512 + pp * 64];
        float t8[8]; float m = INF_F;
#pragma unroll
        for (int i = 0; i < 8; ++i) t8[i] = INF_F;
        // staggered scan: consecutive threads hit different LDS banks
        for (int i = 0; i < 64; ++i) topk_insert(t8, m, src[(i + tid) & 63]);
        float* dst = &cand2[q * 64 + pp * 8];
#pragma unroll
        for (int i = 0; i < 8; ++i) dst[i] = t8[i];
    }
    __syncthreads();

    // ---- stage-2 reduce: 64 -> 8, write partial result ----
    if (tid < QTILE) {
        const float* src = &cand2[tid * 64];
        float t8[8]; float m = INF_F;
#pragma unroll
        for (int i = 0; i < 8; ++i) t8[i] = INF_F;
        for (int i = 0; i < 64; ++i) topk_insert(t8, m, src[(i + tid * 4) & 63]);
        float* dst = part + ((size_t)(q0 + tid) * NSPLIT + split) * KTOP;
#pragma unroll
        for (int i = 0; i < 8; ++i) dst[i] = t8[i];
    }
}

__global__ __launch_bounds__(128) void knn_merge_kernel(
    const float* __restrict__ part,     // [N_QUERY][NSPLIT][KTOP]
    float* __restrict__ out)            // [N_QUERY][KTOP]
{
    int q = blockIdx.x * blockDim.x + threadIdx.x;
    if (q >= N_QUERY) return;

    const float* src = part + (size_t)q * NSPLIT * KTOP;
    float s[8]; float m = INF_F;
#pragma unroll
    for (int i = 0; i < 8; ++i) s[i] = INF_F;
    for (int i = 0; i < NSPLIT * KTOP; ++i) topk_insert(s, m, src[i]);

    // sort ascending: static selection network (no dynamic indexing)
#pragma unroll
    for (int i = 0; i < 8; ++i) {
#pragma unroll
        for (int j = i + 1; j < 8; ++j) {
            if (s[j] < s[i]) { float t = s[i]; s[i] = s[j]; s[j] = t; }
        }
    }
#pragma unroll
    for (int i = 0; i < 8; ++i) out[q * KTOP + i] = s[i];
}

extern "C" void kernel_launch(void* const* d_in, const int* in_sizes, int n_in,
                              void* d_out, int out_size, void* d_ws, size_t ws_size,
                              hipStream_t stream) {
    const float* x    = (const float*)d_in[0];
    const float* base = (const float*)d_in[1];
    float* part = (float*)d_ws;     // needs N_QUERY*NSPLIT*KTOP*4 = 1.6 MB
    float* out  = (float*)d_out;

    dim3 grid1(NSPLIT, NQTILES);    // 50 x 64 = 3200 workgroups
    knn_partial_kernel<<<grid1, BLOCK, 0, stream>>>(x, base, part);

    knn_merge_kernel<<<(N_QUERY + 127) / 128, 128, 0, stream>>>(part, out);
}